// SoftThresholdingOperation_18691697672778
// MI455X (gfx1250) — compile-verified
//
#include <hip/hip_runtime.h>
#include <stdint.h>

// Sparsemax along axis 2 of (8,16,4096,64) f32, MI455X/gfx1250.
//
// Per block: a 4096x8 f32 tile is DMA'd to LDS by the Tensor Data Mover with
// +1-DWORD row padding (row stride 9 words; gcd(9,64)=1 -> conflict-free
// column access). TWO waves per column (16 waves, 512 threads -> 256-VGPR
// budget, no spills; ~96-VGPR allocation lets 2 blocks share a WGP for
// 8 waves/SIMD) keep 64 elements per lane in VGPRs and run the exact
// water-filling Newton iteration for the sparsemax threshold:
//   g(tau) = sum(max(y - tau, 0)) convex piecewise-linear decreasing;
//   tau <- tau + (g(tau)-1)/|{y>tau}| from tau0 = -1 (y = x - max) lands
//   exactly on the root's linear piece in a few steps.
// Cross-wave (s,k) combine uses a double-buffered LDS mailbox: one
// __syncthreads per iteration; phase parity makes slot reuse race-free.
// Output is written in place (same LDS slot, same lane) and stored by a TDM
// tensor_store_from_lds that de-pads via the OOB-drop rule (tile_dim0=9,
// tensor_dim0=8). HBM traffic = load + store only (~256 MiB ~= 11 us).

#define N_AXIS   4096          // reduction axis length (dim 2)
#define D_LAST   64            // innermost dim
#define TJ       8             // columns per block
#define ROW_W    (TJ + 1)      // padded LDS row stride in dwords (9)
#define WPC      2             // waves per column
#define NWAVES   (TJ * WPC)    // 16 waves
#define NTHREADS (NWAVES * 32) // 512 threads
#define NPL      (N_AXIS / (32 * WPC)) // 64 elements per lane

typedef __attribute__((ext_vector_type(4))) unsigned int u32x4;
typedef __attribute__((ext_vector_type(8))) unsigned int u32x8;

__global__ __launch_bounds__(NTHREADS) void sparsemax_tdm_kernel(
    const float* __restrict__ x, float* __restrict__ out) {
  extern __shared__ float smem[]; // tile: N_AXIS*ROW_W floats, + mailbox
  float* redS = smem + N_AXIS * ROW_W; // [2][NWAVES]
  float* redK = redS + 2 * NWAVES;     // [2][NWAVES]

  const int wave = threadIdx.x >> 5;
  const int lane = threadIdx.x & 31;
  const int jj   = wave >> 1;          // column 0..7
  const int half = wave & 1;           // which half of the column
  const int r0   = lane + 32 * half;   // rows r0 + 64*i
  const int jt   = blockIdx.x & (D_LAST / TJ - 1); // 8 j-tiles
  const int bh   = blockIdx.x >> 3;                // (b,h) slice, 0..127
  const int j0   = jt * TJ;
  const uint64_t tileOff =
      ((uint64_t)bh * N_AXIS * D_LAST + (uint64_t)j0) * sizeof(float);

  // ---- TDM load: 4096x8 tile, row stride 64 elems, +1 DW pad per 8 DW ----
  if (wave == 0) {
    const uint64_t gaddr = (uint64_t)(uintptr_t)x + tileOff;
    const uint32_t laddr = (uint32_t)(uintptr_t)&smem[0];
    u32x4 g0; u32x8 g1;
    g0[0] = 1u;                                   // count=1
    g0[1] = laddr;                                // lds_addr
    g0[2] = (uint32_t)gaddr;                      // global_addr lo
    g0[3] = (uint32_t)((gaddr >> 32) & 0x01FFFFFFu) | (2u << 30); // hi|type=2
    g1[0] = (2u << 16)        // data_size = 4 B
          | (1u << 20)        // pad_enable
          | (2u << 22);       // pad_interval code 2 -> every 8 DWORDs, +1 DW
    g1[1] = ((uint32_t)TJ) << 16;       // tensor_dim0 = 8
    g1[2] = ((uint32_t)N_AXIS) << 16;   // tensor_dim1 = 4096
    g1[3] = ((uint32_t)TJ) << 16;       // tile_dim0 = 8
    g1[4] = (uint32_t)N_AXIS;           // tile_dim1 = 4096, tile_dim2 = 0
    g1[5] = (uint32_t)D_LAST;           // tensor_dim0_stride = 64 elems
    g1[6] = 0u;
    g1[7] = 0u;
    asm volatile("tensor_load_to_lds %0, %1" :: "s"(g0), "s"(g1) : "memory");
    __builtin_amdgcn_s_wait_tensorcnt(0);
  }
  __syncthreads();

  // ---- register-resident half-column (one conflict-free LDS read pass) ----
  float y[NPL];
#pragma unroll
  for (int i = 0; i < NPL; ++i)
    y[i] = smem[(r0 + 64 * i) * ROW_W + jj];

  // column max across both waves of the pair (mailbox phase 0)
  float m = y[0];
#pragma unroll
  for (int i = 1; i < NPL; ++i) m = fmaxf(m, y[i]);
#pragma unroll
  for (int off = 16; off > 0; off >>= 1)
    m = fmaxf(m, __shfl_xor(m, off, 32));
  if (lane == 0) redS[wave] = m; // phase-0 slot
  __syncthreads();
  m = fmaxf(m, redS[wave ^ 1]);
#pragma unroll
  for (int i = 0; i < NPL; ++i) y[i] -= m; // shift: y_max = 0, tau* in (-1,0)

  // exact Newton / water-filling iteration on g(tau) = sum((y - tau)_+)
  float tau = -1.0f;
  for (int it = 0; it < 12; ++it) {
    float s = 0.0f, k = 0.0f;
#pragma unroll
    for (int i = 0; i < NPL; ++i) {
      const float t = y[i] - tau;
      if (t > 0.0f) { s += t; k += 1.0f; }
    }
#pragma unroll
    for (int off = 16; off > 0; off >>= 1) {
      s += __shfl_xor(s, off, 32);
      k += __shfl_xor(k, off, 32);
    }
    const int p = (it + 1) & 1; // alternate mailbox phase (race-free reuse)
    if (lane == 0) {
      redS[p * NWAVES + wave] = s;
      redK[p * NWAVES + wave] = k;
    }
    __syncthreads();
    s += redS[p * NWAVES + (wave ^ 1)];
    k += redK[p * NWAVES + (wave ^ 1)];
    tau += (s - 1.0f) / fmaxf(k, 1.0f); // identical in both partner waves
  }

  // in-place output into the padded tile (same lane reads & writes each slot)
#pragma unroll
  for (int i = 0; i < NPL; ++i)
    smem[(r0 + 64 * i) * ROW_W + jj] = fmaxf(y[i] - tau, 0.0f);

  __syncthreads();

  // ---- TDM store: tile_dim0 = 9, tensor_dim0 = 8 -> pad word dropped ----
  if (wave == 0) {
    const uint64_t gaddr = (uint64_t)(uintptr_t)out + tileOff;
    const uint32_t laddr = (uint32_t)(uintptr_t)&smem[0];
    u32x4 g0; u32x8 g1;
    g0[0] = 1u;
    g0[1] = laddr;
    g0[2] = (uint32_t)gaddr;
    g0[3] = (uint32_t)((gaddr >> 32) & 0x01FFFFFFu) | (2u << 30);
    g1[0] = (2u << 16);                 // data_size = 4 B, no pad on store
    g1[1] = ((uint32_t)TJ) << 16;       // tensor_dim0 = 8 (x==8 OOB-drop)
    g1[2] = ((uint32_t)N_AXIS) << 16;   // tensor_dim1 = 4096
    g1[3] = ((uint32_t)ROW_W) << 16;    // tile_dim0 = 9 (walks pad word)
    g1[4] = (uint32_t)N_AXIS;           // tile_dim1 = 4096
    g1[5] = (uint32_t)D_LAST;           // tensor_dim0_stride = 64 elems
    g1[6] = 0u;
    g1[7] = 0u;
    asm volatile("tensor_store_from_lds %0, %1" :: "s"(g0), "s"(g1) : "memory");
    __builtin_amdgcn_s_wait_tensorcnt(0);
  }
}

extern "C" void kernel_launch(void* const* d_in, const int* in_sizes, int n_in,
                              void* d_out, int out_size, void* d_ws, size_t ws_size,
                              hipStream_t stream) {
  const float* x = (const float*)d_in[0];
  float* out = (float*)d_out;

  const int total   = in_sizes[0];               // 8*16*4096*64
  const int nBH     = total / (N_AXIS * D_LAST); // 128
  const int nBlocks = nBH * (D_LAST / TJ);       // 1024

  const size_t shmem =
      ((size_t)N_AXIS * ROW_W + 4 * NWAVES) * sizeof(float); // 147712 B
  sparsemax_tdm_kernel<<<dim3(nBlocks), dim3(NTHREADS), shmem, stream>>>(x, out);
}